// Temporal_78632261255776
// MI455X (gfx1250) — compile-verified
//
#include <hip/hip_runtime.h>

// Problem constants from the reference.
#define T_STEPS 100
#define B_DIM   256
#define F_DIM   4096

// Native clang vector type so __builtin_nontemporal_store lowers to a single
// global_store_b128 with the NT temporal hint (output is 419MB streaming,
// > 2x the 192MB L2 -> bypass/rinse, don't write-allocate).
typedef __attribute__((ext_vector_type(4))) float v4f;

// One thread owns 4 consecutive f-columns of one batch row b.
// It streams T=100 zero float4 rows (fully unrolled, immediate IOFFSETs:
// t*F*4 bytes <= 1.62MB < 2^23, fits the signed 24-bit offset field), then
// patches its <=4 spike elements with 1.0f. Per-wave stores to the same
// address retire in order (CDNA5 ISA 7.3), so the patch lands after the zero.
__global__ __launch_bounds__(256) void spike_onehot_kernel(
    const float* __restrict__ x,
    const float* __restrict__ delays,
    float* __restrict__ out) {
  const int b = blockIdx.y;                                 // 0..255
  const int f = (blockIdx.x * 256 + threadIdx.x) * 4;       // 0..4092 step 4

  const v4f xv = *reinterpret_cast<const v4f*>(x + (size_t)b * F_DIM + f);
  const v4f dv = *reinterpret_cast<const v4f*>(delays + f);

  // st = trunc((1 - x*d) * (T-1)), clipped to [0, T-1]. C-style int cast
  // truncates toward zero, matching the reference's .astype(int32).
  int st0 = (int)((1.0f - xv.x * dv.x) * (float)(T_STEPS - 1));
  int st1 = (int)((1.0f - xv.y * dv.y) * (float)(T_STEPS - 1));
  int st2 = (int)((1.0f - xv.z * dv.z) * (float)(T_STEPS - 1));
  int st3 = (int)((1.0f - xv.w * dv.w) * (float)(T_STEPS - 1));
  st0 = min(max(st0, 0), T_STEPS - 1);
  st1 = min(max(st1, 0), T_STEPS - 1);
  st2 = min(max(st2, 0), T_STEPS - 1);
  st3 = min(max(st3, 0), T_STEPS - 1);

  float* base = out + ((size_t)b * T_STEPS + 0) * F_DIM + f;

  const v4f z = {0.0f, 0.0f, 0.0f, 0.0f};
#pragma unroll
  for (int t = 0; t < T_STEPS; ++t) {
    // global_store_b128 <addr> offset:t*16384 nt
    __builtin_nontemporal_store(
        z, reinterpret_cast<v4f*>(base + (size_t)t * F_DIM));
  }

  // Patch the spike positions (unique addresses per thread; ordered after the
  // zero stores above within this wave).
  base[(size_t)st0 * F_DIM + 0] = 1.0f;
  base[(size_t)st1 * F_DIM + 1] = 1.0f;
  base[(size_t)st2 * F_DIM + 2] = 1.0f;
  base[(size_t)st3 * F_DIM + 3] = 1.0f;
}

extern "C" void kernel_launch(void* const* d_in, const int* in_sizes, int n_in,
                              void* d_out, int out_size, void* d_ws, size_t ws_size,
                              hipStream_t stream) {
  (void)in_sizes; (void)n_in; (void)out_size; (void)d_ws; (void)ws_size;
  const float* x      = (const float*)d_in[0];   // [B, F] f32
  const float* delays = (const float*)d_in[1];   // [F] f32
  float* out          = (float*)d_out;           // [B, T, F] f32

  // 4096 cols / (256 threads * 4 floats) = 4 blocks in x; one y-block per batch row.
  dim3 grid(F_DIM / (256 * 4), B_DIM);
  dim3 block(256);
  spike_onehot_kernel<<<grid, block, 0, stream>>>(x, delays, out);
}